// AdvancedLossFunction_11931419148406
// MI455X (gfx1250) — compile-verified
//
#include <hip/hip_runtime.h>
#include <math.h>

// ---------------------------------------------------------------------------
// AdvancedLossFunction on gfx1250 (MI455X).
// Heavy part: 2 x 4096 x 4096 pairwise-distance matrix, used 3 times
// (row softmax stats, col softmax stats, analytic EMD gradient).
// Distance Gram term computed per 16x16 tile with V_WMMA_F32_16X16X4_F32
// (K=3 padded to 4). Flash-style online softmax, never materializes d.
// Inner loops use raw v_sqrt_f32 / v_rcp_f32 / v_exp_f32 (compute-bound).
// ---------------------------------------------------------------------------

#define INV_TAU   50.0f            // 1 / 0.02
#define K2        72.13475204f     // 1 / (0.02 * ln 2)  -> exp(x/tau) = exp2(x*K2)

typedef __attribute__((ext_vector_type(2))) float v2f;
typedef __attribute__((ext_vector_type(8))) float v8f;

__device__ __forceinline__ float fsqrt(float x)  { return __builtin_amdgcn_sqrtf(x); }
__device__ __forceinline__ float frcp(float x)   { return __builtin_amdgcn_rcpf(x); }
__device__ __forceinline__ float fexp2(float x)  { return __builtin_amdgcn_exp2f(x); }

__device__ __forceinline__ float wave32_sum(float v) {
#pragma unroll
  for (int off = 16; off >= 1; off >>= 1) v += __shfl_xor(v, off, 32);
  return v;
}

// ---------------------------------------------------------------------------
__global__ void k_init(float* acc) {
  if (threadIdx.x < 16) acc[threadIdx.x] = 0.0f;
}

// ---------------------------------------------------------------------------
// Elementwise terms over B*P points: SDF |diff|, BCE, 1-cos(normals), u(1-u).
__global__ void k_elem(const float* __restrict__ sp, const float* __restrict__ st,
                       const float* __restrict__ unc,
                       const float* __restrict__ op, const float* __restrict__ ot,
                       const float* __restrict__ npd, const float* __restrict__ ntg,
                       float* acc, int n) {
  float s_sdf = 0.f, s_bce = 0.f, s_nrm = 0.f, s_unc = 0.f;
  for (int i = blockIdx.x * blockDim.x + threadIdx.x; i < n;
       i += gridDim.x * blockDim.x) {
    s_sdf += fabsf(sp[i] - st[i]);
    float p = op[i], t = ot[i];
    s_bce -= t * __logf(p) + (1.f - t) * __logf(1.f - p);
    float ax = npd[3 * i], ay = npd[3 * i + 1], az = npd[3 * i + 2];
    float bx = ntg[3 * i], by = ntg[3 * i + 1], bz = ntg[3 * i + 2];
    float na = fmaxf(fsqrt(ax * ax + ay * ay + az * az), 1e-8f);
    float nb = fmaxf(fsqrt(bx * bx + by * by + bz * bz), 1e-8f);
    s_nrm += 1.f - (ax * bx + ay * by + az * bz) * frcp(na * nb);
    float u = unc[i];
    s_unc += u * (1.f - u);
  }
  __shared__ float sh[4][8];
  int lane = threadIdx.x & 31, w = threadIdx.x >> 5;
  float v0 = wave32_sum(s_sdf), v1 = wave32_sum(s_bce);
  float v2 = wave32_sum(s_nrm), v3 = wave32_sum(s_unc);
  if (lane == 0) { sh[0][w] = v0; sh[1][w] = v1; sh[2][w] = v2; sh[3][w] = v3; }
  __syncthreads();
  if (threadIdx.x == 0) {
    float t0 = 0, t1 = 0, t2 = 0, t3 = 0;
    int nw = blockDim.x >> 5;
    for (int k = 0; k < nw; ++k) { t0 += sh[0][k]; t1 += sh[1][k]; t2 += sh[2][k]; t3 += sh[3][k]; }
    atomicAdd(&acc[0], t0); atomicAdd(&acc[1], t1);
    atomicAdd(&acc[2], t2); atomicAdd(&acc[3], t3);
  }
}

// ---------------------------------------------------------------------------
__global__ void k_norm2(const float* __restrict__ pts, float* __restrict__ out, int n) {
  int i = blockIdx.x * blockDim.x + threadIdx.x;
  if (i < n) {
    float x = pts[3 * i], y = pts[3 * i + 1], z = pts[3 * i + 2];
    out[i] = x * x + y * y + z * z;
  }
}

// ---------------------------------------------------------------------------
// Row stats of d(A_i, B_j): per-row min (chamfer), Z, S (flash softmax of -d/tau).
// One wave per 16-row strip. 16x16 dot tile via V_WMMA_F32_16X16X4_F32.
// A fragment (16x4, 32-bit): lanes 0-15 rows M=0..15 with VGPR0=K0, VGPR1=K1;
// lanes 16-31 same rows with VGPR0=K2, VGPR1=K3 (ISA 7.12.2). B assumed same striping.
// C/D: VGPR r, lanes 0-15 -> (M=r, N=lane); lanes 16-31 -> (M=8+r, N=lane-16).
__global__ void k_stats(const float* __restrict__ Apts, const float* __restrict__ Bpts,
                        const float* __restrict__ An2,
                        float* __restrict__ oMin, float* __restrict__ oZ,
                        float* __restrict__ oS,
                        int nA, int nB, float* chamAcc, float* emdAcc) {
  int lane = threadIdx.x & 31;
  int strip = blockIdx.x * (blockDim.x >> 5) + (threadIdx.x >> 5);
  if (strip >= (nA >> 4)) return;            // wave-uniform exit (EXEC stays all-ones)
  int l16 = lane & 15, half = lane >> 4;
  int i0 = strip << 4;

  int rowA = i0 + l16;
  float ax = Apts[3 * rowA], ay = Apts[3 * rowA + 1], az = Apts[3 * rowA + 2];
  v2f afrag;
  afrag[0] = half ? az : ax;                 // K0 | K2
  afrag[1] = half ? 0.f : ay;                // K1 | K3(=pad 0)

  float na2[8], m[8], Z[8], S[8];
#pragma unroll
  for (int r = 0; r < 8; ++r) {
    na2[r] = An2[i0 + half * 8 + r];
    m[r] = 3.0e38f; Z[r] = 0.f; S[r] = 0.f;
  }

  int nTiles = nB >> 4;
  for (int jt = 0; jt < nTiles; ++jt) {
    int j = (jt << 4) + l16;
    float bx = Bpts[3 * j], by = Bpts[3 * j + 1], bz = Bpts[3 * j + 2];
    float nb2 = bx * bx + by * by + bz * bz;
    v2f bfrag;
    bfrag[0] = half ? bz : bx;
    bfrag[1] = half ? 0.f : by;
    v8f dot = {};
    dot = __builtin_amdgcn_wmma_f32_16x16x4_f32(false, afrag, false, bfrag,
                                                (short)0, dot, false, false);
#pragma unroll
    for (int r = 0; r < 8; ++r) {
      float d2 = na2[r] + nb2 - 2.f * dot[r];
      float d = fsqrt(fmaxf(d2, 1e-12f));
      // single-exp online softmax update:
      // if d < m : rescale factor = t, new term = 1 ; else factor = 1, term = t
      float t = fexp2(-fabsf(d - m[r]) * K2);
      bool nmin = d < m[r];
      float f = nmin ? t : 1.f;
      float e = nmin ? 1.f : t;
      Z[r] = Z[r] * f + e;
      S[r] = S[r] * f + e * d;
      m[r] = fminf(m[r], d);
    }
  }
  // merge the 16 per-lane partial softmax states within each half-wave
#pragma unroll
  for (int r = 0; r < 8; ++r) {
#pragma unroll
    for (int off = 1; off < 16; off <<= 1) {
      float m2 = __shfl_xor(m[r], off, 32);
      float Z2 = __shfl_xor(Z[r], off, 32);
      float S2 = __shfl_xor(S[r], off, 32);
      float nm = fminf(m[r], m2);
      float f1 = fexp2((nm - m[r]) * K2);
      float f2 = fexp2((nm - m2) * K2);
      Z[r] = Z[r] * f1 + Z2 * f2;
      S[r] = S[r] * f1 + S2 * f2;
      m[r] = nm;
    }
  }
  if (l16 == 0) {
    float cham = 0.f, emd = 0.f;
#pragma unroll
    for (int r = 0; r < 8; ++r) {
      int row = i0 + half * 8 + r;
      oMin[row] = m[r]; oZ[row] = Z[r]; oS[row] = S[r];
      cham += m[r];
      emd += S[r] * frcp(Z[r]);
    }
    atomicAdd(chamAcc, cham);
    atomicAdd(emdAcc, emd);
  }
}

// ---------------------------------------------------------------------------
// Analytic EMD gradient wrt pred points + | ||g|| - 1 | accumulation.
// g_i = sum_j [a*r_ij(1-(d-Er_i)/tau) + b*c_ij(1-(d-Ec_j)/tau)] (a_i-b_j)/d
__global__ void k_grad(const float* __restrict__ Apts, const float* __restrict__ Bpts,
                       const float* __restrict__ An2,
                       const float* __restrict__ rMin, const float* __restrict__ rZ,
                       const float* __restrict__ rS,
                       const float* __restrict__ cMin, const float* __restrict__ cZ,
                       const float* __restrict__ cS,
                       int nA, int nB, float alpha, float beta, float* gpAcc) {
  int lane = threadIdx.x & 31;
  int strip = blockIdx.x * (blockDim.x >> 5) + (threadIdx.x >> 5);
  if (strip >= (nA >> 4)) return;
  int l16 = lane & 15, half = lane >> 4;
  int i0 = strip << 4;

  int rowA = i0 + l16;
  float fax = Apts[3 * rowA], fay = Apts[3 * rowA + 1], faz = Apts[3 * rowA + 2];
  v2f afrag;
  afrag[0] = half ? faz : fax;
  afrag[1] = half ? 0.f : fay;

  float arx[8], ary[8], arz[8], na2[8], rm[8], rzi[8], re[8];
  float gx[8], gy[8], gz[8];
#pragma unroll
  for (int r = 0; r < 8; ++r) {
    int row = i0 + half * 8 + r;
    arx[r] = Apts[3 * row]; ary[r] = Apts[3 * row + 1]; arz[r] = Apts[3 * row + 2];
    na2[r] = An2[row];
    rm[r] = rMin[row];
    rzi[r] = frcp(rZ[row]);
    re[r] = rS[row] * rzi[r];
    gx[r] = 0.f; gy[r] = 0.f; gz[r] = 0.f;
  }

  int nTiles = nB >> 4;
  for (int jt = 0; jt < nTiles; ++jt) {
    int j = (jt << 4) + l16;
    float bx = Bpts[3 * j], by = Bpts[3 * j + 1], bz = Bpts[3 * j + 2];
    float nb2 = bx * bx + by * by + bz * bz;
    float cm = cMin[j];
    float czi = frcp(cZ[j]);
    float ce = cS[j] * czi;
    v2f bfrag;
    bfrag[0] = half ? bz : bx;
    bfrag[1] = half ? 0.f : by;
    v8f dot = {};
    dot = __builtin_amdgcn_wmma_f32_16x16x4_f32(false, afrag, false, bfrag,
                                                (short)0, dot, false, false);
#pragma unroll
    for (int r = 0; r < 8; ++r) {
      float d2 = na2[r] + nb2 - 2.f * dot[r];
      float d = fsqrt(fmaxf(d2, 1e-12f));
      float er = fexp2((rm[r] - d) * K2) * rzi[r];   // r_ij
      float ec = fexp2((cm - d) * K2) * czi;         // c_ij
      float coef = alpha * er * (1.f - (d - re[r]) * INV_TAU)
                 + beta  * ec * (1.f - (d - ce)   * INV_TAU);
      float w = coef * frcp(d);
      gx[r] += w * (arx[r] - bx);
      gy[r] += w * (ary[r] - by);
      gz[r] += w * (arz[r] - bz);
    }
  }
  // sum partial gradients across the 16 lanes of each half
#pragma unroll
  for (int r = 0; r < 8; ++r) {
#pragma unroll
    for (int off = 1; off < 16; off <<= 1) {
      gx[r] += __shfl_xor(gx[r], off, 32);
      gy[r] += __shfl_xor(gy[r], off, 32);
      gz[r] += __shfl_xor(gz[r], off, 32);
    }
  }
  if (l16 == 0) {
    float s = 0.f;
#pragma unroll
    for (int r = 0; r < 8; ++r) {
      float nrm = fsqrt(gx[r] * gx[r] + gy[r] * gy[r] + gz[r] * gz[r]);
      s += fabsf(nrm - 1.f);
    }
    atomicAdd(gpAcc, s);
  }
}

// ---------------------------------------------------------------------------
__global__ void k_final(const float* __restrict__ acc, float* __restrict__ out,
                        int BP, int BN, int BM) {
  if (blockIdx.x == 0 && threadIdx.x == 0) {
    float invBP = 1.f / (float)BP;
    float sdf_loss = acc[0] * invBP;               // weights normalize to mean 1
    float bce = acc[1] * invBP;
    float pt = __expf(-bce);
    float occ = 0.75f * (1.f - pt) * (1.f - pt) * bce;
    float nrm = acc[2] * invBP;
    float unc = acc[3] * invBP;
    float cham = acc[4] / (float)BN + acc[5] / (float)BM;
    float emd  = acc[6] / (float)BN + acc[7] / (float)BM;
    float gp   = acc[8] / (float)BN;
    out[0] = 1.0f * sdf_loss + 1.0f * occ + 0.1f * nrm + 1.0f * cham
           + 0.25f * emd + 0.05f * gp + 0.1f * unc;
  }
}

// ---------------------------------------------------------------------------
extern "C" void kernel_launch(void* const* d_in, const int* in_sizes, int n_in,
                              void* d_out, int out_size, void* d_ws, size_t ws_size,
                              hipStream_t stream) {
  (void)n_in; (void)out_size; (void)ws_size;
  const float* sdf_pred   = (const float*)d_in[0];
  const float* sdf_target = (const float*)d_in[1];
  const float* uncert     = (const float*)d_in[2];
  const float* occ_pred   = (const float*)d_in[3];
  const float* occ_target = (const float*)d_in[4];
  const float* nrm_pred   = (const float*)d_in[5];
  const float* nrm_target = (const float*)d_in[6];
  const float* pc_pred    = (const float*)d_in[7];
  const float* pc_target  = (const float*)d_in[8];

  const int B  = 2;
  const int BP = in_sizes[0];
  const int N  = in_sizes[7] / (B * 3);
  const int M  = in_sizes[8] / (B * 3);

  float* ws     = (float*)d_ws;
  float* acc    = ws;                 // 16 accumulators
  float* predN2 = ws + 16;            // B*N
  float* tgtN2  = predN2 + B * N;     // B*M
  float* rMin   = tgtN2 + B * M;      // B*N row stats
  float* rZ     = rMin + B * N;
  float* rS     = rZ + B * N;
  float* cMin   = rS + B * N;         // B*M col stats
  float* cZ     = cMin + B * M;
  float* cS     = cZ + B * M;

  k_init<<<1, 32, 0, stream>>>(acc);
  k_elem<<<256, 256, 0, stream>>>(sdf_pred, sdf_target, uncert, occ_pred,
                                  occ_target, nrm_pred, nrm_target, acc, BP);
  k_norm2<<<(B * N + 255) / 256, 256, 0, stream>>>(pc_pred, predN2, B * N);
  k_norm2<<<(B * M + 255) / 256, 256, 0, stream>>>(pc_target, tgtN2, B * M);

  const int TPB = 256, WPB = TPB / 32;
  for (int b = 0; b < B; ++b) {
    const float* Ap = pc_pred + (size_t)b * N * 3;
    const float* Bp = pc_target + (size_t)b * M * 3;
    int blocksR = (N / 16 + WPB - 1) / WPB;
    k_stats<<<blocksR, TPB, 0, stream>>>(Ap, Bp, predN2 + b * N,
                                         rMin + b * N, rZ + b * N, rS + b * N,
                                         N, M, acc + 4, acc + 6);
    int blocksC = (M / 16 + WPB - 1) / WPB;
    k_stats<<<blocksC, TPB, 0, stream>>>(Bp, Ap, tgtN2 + b * M,
                                         cMin + b * M, cZ + b * M, cS + b * M,
                                         M, N, acc + 5, acc + 7);
  }
  float alpha = 1.f / (float)(B * N), beta = 1.f / (float)(B * M);
  for (int b = 0; b < B; ++b) {
    int blocksG = (N / 16 + WPB - 1) / WPB;
    k_grad<<<blocksG, TPB, 0, stream>>>(pc_pred + (size_t)b * N * 3,
                                        pc_target + (size_t)b * M * 3,
                                        predN2 + b * N,
                                        rMin + b * N, rZ + b * N, rS + b * N,
                                        cMin + b * M, cZ + b * M, cS + b * M,
                                        N, M, alpha, beta, acc + 8);
  }
  k_final<<<1, 32, 0, stream>>>(acc, (float*)d_out, BP, B * N, B * M);
}